// GCN_36352603193802
// MI455X (gfx1250) — compile-verified
//
#include <hip/hip_runtime.h>
#include <hip/hip_bf16.h>

typedef __attribute__((ext_vector_type(2))) float v2f;
typedef __attribute__((ext_vector_type(8))) float v8f;

#define N_NODES 100000
#define N_EDGES 1600000
#define KDIM    128   // input/hidden feature width (= GEMM K)
#define M_TILES (N_NODES / 16)   // 6250, exact

__device__ __forceinline__ int lower_bound_i(const int* __restrict__ a, int n, int key) {
    int lo = 0, hi = n;
    while (lo < hi) {
        int mid = (lo + hi) >> 1;
        if (a[mid] < key) lo = mid + 1; else hi = mid;
    }
    return lo;
}

// dinv[i] = deg>0 ? rsqrt(deg) : 0 ; deg from span length in sorted edge_row
__global__ void gcn_dinv_kernel(const int* __restrict__ erow, float* __restrict__ dinv) {
    int i = blockIdx.x * blockDim.x + threadIdx.x;
    if (i >= N_NODES) return;
    int s = lower_bound_i(erow, N_EDGES, i);
    int e = lower_bound_i(erow, N_EDGES, i + 1);
    int cnt = e - s;
    dinv[i] = (cnt > 0) ? rsqrtf((float)cnt) : 0.0f;
}

// out[M,NCOL] = A[M,128] @ W[128,NCOL] + bias, via V_WMMA_F32_16X16X4_F32.
// 4 waves per block, one 16-row M-tile per wave, full N per wave.
// W is staged in LDS K-pair-interleaved: wlds[(k>>1)*2*NCOL + 2*n + (k&1)] = W[k][n]
// so each B fragment {W[kb][n], W[kb+1][n]} is one aligned ds_load_b64.
template <int NCOL>
__global__ __launch_bounds__(128) void gcn_gemm_wmma(const float* __restrict__ A,
                                                     const float* __restrict__ W,
                                                     const float* __restrict__ bias,
                                                     float* __restrict__ out) {
    constexpr int NT = NCOL / 16;                 // N-tiles per wave
    __shared__ float wlds[KDIM * NCOL];           // 64KB (NCOL=128) / 32KB (NCOL=64)

    // Stage W into LDS, pair-interleaving consecutive K rows.
    for (int idx = threadIdx.x; idx < KDIM * NCOL; idx += blockDim.x) {
        int k = idx / NCOL;
        int n = idx % NCOL;
        wlds[(k >> 1) * (2 * NCOL) + 2 * n + (k & 1)] = W[idx];
    }
    __syncthreads();

    const int wave = threadIdx.x >> 5;
    const int lane = threadIdx.x & 31;
    const int lo   = lane & 15;    // M (for A/C) or N (for B/C) index within tile
    const int hi   = lane >> 4;    // selects K half / M half per ISA layout

    const int tile = blockIdx.x * 4 + wave;       // wave-uniform
    if (tile >= M_TILES) return;                  // uniform exit (after barrier)

    const float* arow = A + (size_t)(tile * 16 + lo) * KDIM;

    v8f zero = {};
    v8f acc[NT];
#pragma unroll
    for (int t = 0; t < NT; ++t) acc[t] = zero;

    for (int k = 0; k < KDIM; k += 4) {
        const int kb = k + 2 * hi;                 // even
        // A 16x4 frag: VGPR v holds A[lo][kb+v]  (one aligned b64 load)
        v2f a = *(const v2f*)(arow + kb);
        const float* wrow = &wlds[(kb >> 1) * (2 * NCOL)];
#pragma unroll
        for (int t = 0; t < NT; ++t) {
            // B 4x16 frag: VGPR v holds B[kb+v][t*16+lo] (one aligned ds_load_b64)
            v2f b = *(const v2f*)(wrow + 2 * (t * 16 + lo));
            acc[t] = __builtin_amdgcn_wmma_f32_16x16x4_f32(
                false, a, false, b, (short)0, acc[t], false, false);
        }
    }

    // C layout: VGPR v = C[M = v + 8*hi][N = lo]
#pragma unroll
    for (int t = 0; t < NT; ++t) {
        float bv = bias[t * 16 + lo];
#pragma unroll
        for (int v = 0; v < 8; ++v) {
            int m = tile * 16 + v + 8 * hi;
            out[(size_t)m * NCOL + t * 16 + lo] = acc[t][v] + bv;
        }
    }
}

// out[i,:] = (relu of) sum_{e in row i} dinv[i]*dinv[col[e]] * h[col[e],:]
// one block per node, one thread per feature; h rows hit L2 (h fits in 192MB L2).
// Prefetch next gathered row (global_prefetch_b8) to overlap the gather chain.
template <int F, bool RELU>
__global__ __launch_bounds__(F) void gcn_spmm(const int* __restrict__ erow,
                                              const int* __restrict__ ecol,
                                              const float* __restrict__ dinv,
                                              const float* __restrict__ h,
                                              float* __restrict__ out) {
    const int i = blockIdx.x;
    const int t = threadIdx.x;
    const int s = lower_bound_i(erow, N_EDGES, i);
    const int e = lower_bound_i(erow, N_EDGES, i + 1);
    const float di = dinv[i];
    float acc = 0.0f;
    for (int ed = s; ed < e; ++ed) {
        int c = ecol[ed];                   // scalar broadcast across block
        if (ed + 1 < e) {
            int cn = ecol[ed + 1];
            __builtin_prefetch(&h[(size_t)cn * F + t], 0, 0);
        }
        float v = di * dinv[c];
        acc = fmaf(v, h[(size_t)c * F + t], acc);
    }
    if (RELU) acc = fmaxf(acc, 0.0f);
    out[(size_t)i * F + t] = acc;
}

// Second SpMM fused with 64-wide log_softmax. 64 threads per node.
__global__ __launch_bounds__(64) void gcn_spmm_lsm(const int* __restrict__ erow,
                                                   const int* __restrict__ ecol,
                                                   const float* __restrict__ dinv,
                                                   const float* __restrict__ h,
                                                   float* __restrict__ out) {
    const int i = blockIdx.x;
    const int t = threadIdx.x;
    __shared__ float red[64];

    const int s = lower_bound_i(erow, N_EDGES, i);
    const int e = lower_bound_i(erow, N_EDGES, i + 1);
    const float di = dinv[i];
    float acc = 0.0f;
    for (int ed = s; ed < e; ++ed) {
        int c = ecol[ed];
        if (ed + 1 < e) {
            int cn = ecol[ed + 1];
            __builtin_prefetch(&h[(size_t)cn * 64 + t], 0, 0);
        }
        float v = di * dinv[c];
        acc = fmaf(v, h[(size_t)c * 64 + t], acc);
    }

    // max-reduce over 64
    red[t] = acc; __syncthreads();
    for (int st = 32; st > 0; st >>= 1) {
        if (t < st) red[t] = fmaxf(red[t], red[t + st]);
        __syncthreads();
    }
    const float m = red[0];
    __syncthreads();

    // sum exp(acc - m)
    red[t] = __expf(acc - m); __syncthreads();
    for (int st = 32; st > 0; st >>= 1) {
        if (t < st) red[t] += red[t + st];
        __syncthreads();
    }
    out[(size_t)i * 64 + t] = acc - m - __logf(red[0]);
}

extern "C" void kernel_launch(void* const* d_in, const int* in_sizes, int n_in,
                              void* d_out, int out_size, void* d_ws, size_t ws_size,
                              hipStream_t stream) {
    const float* x    = (const float*)d_in[0];   // [100000,128]
    const int*   erow = (const int*)  d_in[1];   // [1.6M] sorted
    const int*   ecol = (const int*)  d_in[2];   // [1.6M]
    const float* W0   = (const float*)d_in[3];   // [128,128]
    const float* b0   = (const float*)d_in[4];   // [128]
    const float* W1   = (const float*)d_in[5];   // [128,64]
    const float* b1   = (const float*)d_in[6];   // [64]
    float* out = (float*)d_out;                  // [100000,64]

    // Workspace layout: dinv | h0 | t0 ; h1 aliases h0 (dead after SpMM0)
    char*  ws   = (char*)d_ws;
    float* dinv = (float*)ws;                                   // 100000 (pad to 102400)
    float* h0   = (float*)(ws + (size_t)102400 * sizeof(float));// [100000,128]
    float* t0   = h0 + (size_t)N_NODES * 128;                   // [100000,128]
    float* h1   = h0;                                           // [100000,64] reuse

    gcn_dinv_kernel<<<(N_NODES + 255) / 256, 256, 0, stream>>>(erow, dinv);

    // layer 0: linear -> SpMM -> relu
    gcn_gemm_wmma<128><<<(M_TILES + 3) / 4, 128, 0, stream>>>(x, W0, b0, h0);
    gcn_spmm<128, true><<<N_NODES, 128, 0, stream>>>(erow, ecol, dinv, h0, t0);

    // layer 1: linear -> SpMM -> log_softmax
    gcn_gemm_wmma<64><<<(M_TILES + 3) / 4, 128, 0, stream>>>(t0, W1, b1, h1);
    gcn_spmm_lsm<<<N_NODES, 64, 0, stream>>>(erow, ecol, dinv, h1, out);
}